// LocalPoolPointnet_34626026340939
// MI455X (gfx1250) — compile-verified
//
#include <hip/hip_runtime.h>

#define B_     16
#define T_     8192
#define N_     (B_ * T_)        // 131072 points
#define H_     128
#define K3_    384
#define RESO_  64
#define R2_    (RESO_ * RESO_)  // 4096
#define CELLS_ (B_ * R2_)       // 65536
#define NB_    5
#define ROWS_  64               // rows per workgroup
#define LDX_   388              // padded LDS stride (mod 64 == 4 -> conflict-free; x4 bytes 16B-aligned)
#define LDT_   132              // padded LDS stride for 128-wide tile

typedef __attribute__((ext_vector_type(2))) float v2f;
typedef __attribute__((ext_vector_type(8))) float v8f;

static __device__ inline v8f zero8() { v8f z = {}; return z; }

// Monotonic float -> uint key for atomicMax-based float max (handles negatives).
static __device__ inline unsigned fkey(float f) {
    unsigned u = __float_as_uint(f);
    return (u & 0x80000000u) ? ~u : (u | 0x80000000u);
}
static __device__ inline float unkey(unsigned k) {
    unsigned u = (k & 0x80000000u) ? (k ^ 0x80000000u) : ~k;
    return __uint_as_float(u);
}

static __device__ inline int cell_of(float a, float c) {
    const float inv = 1.0f / (1.0f + 0.1f + 1e-3f);
    float u = fminf(fmaxf(a * inv + 0.5f, 0.0f), 1.0f - 1e-5f);
    float v = fminf(fmaxf(c * inv + 0.5f, 0.0f), 1.0f - 1e-5f);
    int xi = (int)(u * (float)RESO_);
    int yi = (int)(v * (float)RESO_);
    return xi + RESO_ * yi;
}

// ---------------- index precompute: 3 plane cell indices per point ----------
__global__ void idx_kernel(const float* __restrict__ p, int* __restrict__ idx) {
    int t = blockIdx.x * blockDim.x + threadIdx.x;
    if (t >= N_) return;
    float c0 = p[t * 3 + 0] * 0.125f;   // / SCALE
    float c1 = p[t * 3 + 1] * 0.125f;
    float c2 = p[t * 3 + 2] * 0.125f;
    int b = t / T_;
    idx[0 * N_ + t] = b * R2_ + cell_of(c0, c2);  // xz
    idx[1 * N_ + t] = b * R2_ + cell_of(c0, c1);  // xy
    idx[2 * N_ + t] = b * R2_ + cell_of(c1, c2);  // yz
}

// ---------------- generic zero fill ----------------------------------------
__global__ void zero_kernel(unsigned* __restrict__ buf, long long n) {
    long long i = (long long)blockIdx.x * blockDim.x + threadIdx.x;
    long long stride = (long long)gridDim.x * blockDim.x;
    for (; i < n; i += stride) buf[i] = 0u;
}

// ---------------- scatter-max (segment_max) into per-plane cell buffers -----
__global__ void scatter_max_kernel(const float* __restrict__ net,
                                   const int* __restrict__ idx,
                                   unsigned* __restrict__ cell) {
    int g = blockIdx.x * blockDim.x + threadIdx.x;  // over N_*H_
    if (g >= N_ * H_) return;
    int t = g >> 7, ch = g & 127;
    unsigned key = fkey(net[g]);
#pragma unroll
    for (int k = 0; k < 3; ++k) {
        int ci = idx[k * N_ + t];
        atomicMax(&cell[((size_t)k * CELLS_ + (size_t)ci) * H_ + ch], key);
    }
}

// ---------------- mean over T (hierarchical: 256-row partials) --------------
__global__ void mean_kernel(const float* __restrict__ net, float* __restrict__ meanbuf) {
    int ch = threadIdx.x;             // 0..127
    const int segs = T_ / 256;
    int b = blockIdx.x / segs, seg = blockIdx.x % segs;
    const float* base = net + ((size_t)(b * T_ + seg * 256)) * H_;
    float s = 0.f;
    for (int r = 0; r < 256; ++r) s += base[(size_t)r * H_ + ch];
    atomicAdd(&meanbuf[b * H_ + ch], s);
}

// ---------------- fused ResnetBlockFC, WMMA f32 16x16x4 ---------------------
// Workgroup: 256 threads (8 waves), computes 64 rows x 128 cols of output.
// Wave w owns N-tile w; each wave iterates 4 M-tiles.
// GEMM1 (relu(x)@W0) and GEMM3 (x@Ws) are fused into one K3 k-loop so the
// A operand is read from LDS once: per k-step -> 4 A loads + 2 B loads, 8 WMMAs.
template <bool FIRST>
__global__ void resblock_kernel(const float* __restrict__ in,     // FIRST: p(N,3) else net(N,128)
                                const float* __restrict__ Wpos,   // (3,384) | unused
                                const float* __restrict__ bpos,   // (384)   | unused
                                const unsigned* __restrict__ cell,// [3][CELLS_][128] keys | unused
                                const int* __restrict__ idx,      // [3][N]  | unused
                                const float* __restrict__ meanbuf,// [B][128] sums | unused
                                const float* __restrict__ W0, const float* __restrict__ b0,
                                const float* __restrict__ W1, const float* __restrict__ b1,
                                const float* __restrict__ Ws,
                                float* __restrict__ out) {
    extern __shared__ float smem[];
    float* xs = smem;                   // ROWS_ x LDX_
    float* ts = smem + ROWS_ * LDX_;    // ROWS_ x LDT_
    const int row0 = blockIdx.x * ROWS_;

    // ---- build concat input tile x = [net | sum-of-plane max-gather | mean] ----
    if (FIRST) {
        // x = p @ fc_pos_W + b, computed in-register (K=3), vectorized over cols
        for (int f = threadIdx.x; f < ROWS_ * (K3_ / 4); f += blockDim.x) {
            int r = f / (K3_ / 4), c4 = (f % (K3_ / 4)) << 2;
            int row = row0 + r;
            float p0 = in[row * 3 + 0], p1 = in[row * 3 + 1], p2 = in[row * 3 + 2];
            float4 w0 = *(const float4*)(Wpos + c4);
            float4 w1 = *(const float4*)(Wpos + K3_ + c4);
            float4 w2 = *(const float4*)(Wpos + 2 * K3_ + c4);
            float4 bb = *(const float4*)(bpos + c4);
            float4 v;
            v.x = p0 * w0.x + p1 * w1.x + p2 * w2.x + bb.x;
            v.y = p0 * w0.y + p1 * w1.y + p2 * w2.y + bb.y;
            v.z = p0 * w0.z + p1 * w1.z + p2 * w2.z + bb.z;
            v.w = p0 * w0.w + p1 * w1.w + p2 * w2.w + bb.w;
            *(float4*)(xs + r * LDX_ + c4) = v;
        }
    } else {
        // region 0: net, float4 loads/stores (global_load_b128 / ds_store_b128)
        for (int f = threadIdx.x; f < ROWS_ * (H_ / 4); f += blockDim.x) {
            int r = f >> 5, c4 = (f & 31) << 2;
            float4 v = *(const float4*)(in + (size_t)(row0 + r) * H_ + c4);
            *(float4*)(xs + r * LDX_ + c4) = v;
        }
        // region 1: pooled = sum over 3 planes of gathered per-cell max
        for (int f = threadIdx.x; f < ROWS_ * H_; f += blockDim.x) {
            int r = f >> 7, ch = f & 127;
            int row = row0 + r;
            float s = 0.f;
#pragma unroll
            for (int k = 0; k < 3; ++k) {
                int ci = idx[k * N_ + row];
                s += unkey(cell[((size_t)k * CELLS_ + (size_t)ci) * H_ + ch]);
            }
            xs[r * LDX_ + H_ + ch] = s;
        }
        // region 2: broadcast mean
        for (int f = threadIdx.x; f < ROWS_ * H_; f += blockDim.x) {
            int r = f >> 7, ch = f & 127;
            int b = (row0 + r) / T_;
            xs[r * LDX_ + 2 * H_ + ch] = meanbuf[b * H_ + ch] * (1.0f / (float)T_);
        }
    }
    __syncthreads();

    const int lane = threadIdx.x & 31;
    const int wv   = threadIdx.x >> 5;    // N-tile
    const int col  = wv * 16 + (lane & 15);
    const int kh   = (lane >> 4) << 1;    // 0 or 2 (K half per ISA A/B layout)
    const int rofs = (lane >> 4) << 3;    // C: M = vgpr + 8*(lane>=16)

    // ---- fused GEMM1 + GEMM3 over K3: acc0 = relu(x)@W0 ; accS = x@Ws ----
    v8f acc0[4], accS[4];
#pragma unroll
    for (int m = 0; m < 4; ++m) { acc0[m] = zero8(); accS[m] = zero8(); }
    for (int k0 = 0; k0 < K3_; k0 += 4) {
        v2f bf0, bfs;
        bf0.x = W0[(k0 + kh) * H_ + col];
        bf0.y = W0[(k0 + kh + 1) * H_ + col];
        bfs.x = Ws[(k0 + kh) * H_ + col];
        bfs.y = Ws[(k0 + kh + 1) * H_ + col];
#pragma unroll
        for (int m = 0; m < 4; ++m) {
            int r = m * 16 + (lane & 15);
            v2f ax, ar;
            ax.x = xs[r * LDX_ + k0 + kh];
            ax.y = xs[r * LDX_ + k0 + kh + 1];
            ar.x = fmaxf(ax.x, 0.f);
            ar.y = fmaxf(ax.y, 0.f);
            acc0[m] = __builtin_amdgcn_wmma_f32_16x16x4_f32(false, ar, false, bf0,
                                                            (short)0, acc0[m], false, false);
            accS[m] = __builtin_amdgcn_wmma_f32_16x16x4_f32(false, ax, false, bfs,
                                                            (short)0, accS[m], false, false);
        }
    }
    float bias0 = b0[col];
#pragma unroll
    for (int m = 0; m < 4; ++m)
#pragma unroll
        for (int r = 0; r < 8; ++r)
            ts[(m * 16 + rofs + r) * LDT_ + col] = fmaxf(acc0[m][r] + bias0, 0.f);
    __syncthreads();

    // ---- GEMM2: accS += relu(net0) @ W1   (A from ts, already relu'd) ----
    for (int k0 = 0; k0 < H_; k0 += 4) {
        v2f bf;
        bf.x = W1[(k0 + kh) * H_ + col];
        bf.y = W1[(k0 + kh + 1) * H_ + col];
#pragma unroll
        for (int m = 0; m < 4; ++m) {
            int r = m * 16 + (lane & 15);
            v2f af;
            af.x = ts[r * LDT_ + k0 + kh];
            af.y = ts[r * LDT_ + k0 + kh + 1];
            accS[m] = __builtin_amdgcn_wmma_f32_16x16x4_f32(false, af, false, bf,
                                                            (short)0, accS[m], false, false);
        }
    }
    float bias1 = b1[col];
#pragma unroll
    for (int m = 0; m < 4; ++m)
#pragma unroll
        for (int r = 0; r < 8; ++r)
            out[(size_t)(row0 + m * 16 + rofs + r) * H_ + col] = accS[m][r] + bias1;
}

// ---------------- c = net @ fc_c_W + fc_c_b (WMMA, K=128) -------------------
__global__ void fcc_kernel(const float* __restrict__ in, const float* __restrict__ W,
                           const float* __restrict__ bias, float* __restrict__ out) {
    extern __shared__ float smem[];
    float* xs = smem;  // ROWS_ x LDT_
    const int row0 = blockIdx.x * ROWS_;
    for (int f = threadIdx.x; f < ROWS_ * (H_ / 4); f += blockDim.x) {
        int r = f >> 5, c4 = (f & 31) << 2;
        float4 v = *(const float4*)(in + (size_t)(row0 + r) * H_ + c4);
        *(float4*)(xs + r * LDT_ + c4) = v;
    }
    __syncthreads();

    const int lane = threadIdx.x & 31;
    const int wv   = threadIdx.x >> 5;
    const int col  = wv * 16 + (lane & 15);
    const int kh   = (lane >> 4) << 1;
    const int rofs = (lane >> 4) << 3;

    v8f acc[4];
#pragma unroll
    for (int m = 0; m < 4; ++m) acc[m] = zero8();
    for (int k0 = 0; k0 < H_; k0 += 4) {
        v2f bf;
        bf.x = W[(k0 + kh) * H_ + col];
        bf.y = W[(k0 + kh + 1) * H_ + col];
#pragma unroll
        for (int m = 0; m < 4; ++m) {
            int r = m * 16 + (lane & 15);
            v2f af;
            af.x = xs[r * LDT_ + k0 + kh];
            af.y = xs[r * LDT_ + k0 + kh + 1];
            acc[m] = __builtin_amdgcn_wmma_f32_16x16x4_f32(false, af, false, bf,
                                                           (short)0, acc[m], false, false);
        }
    }
    float bb = bias[col];
#pragma unroll
    for (int m = 0; m < 4; ++m)
#pragma unroll
        for (int r = 0; r < 8; ++r)
            out[(size_t)(row0 + m * 16 + rofs + r) * H_ + col] = acc[m][r] + bb;
}

// ---------------- scatter-mean numerators + counts --------------------------
__global__ void scatter_add_kernel(const float* __restrict__ c, const int* __restrict__ idx,
                                   float* __restrict__ out, float* __restrict__ cnt) {
    int g = blockIdx.x * blockDim.x + threadIdx.x;  // over N_*H_
    if (g >= N_ * H_) return;
    int t = g >> 7, ch = g & 127;
    float v = c[g];
    int b = t / T_;
#pragma unroll
    for (int k = 0; k < 3; ++k) {
        int ci = idx[k * N_ + t];         // = b*R2 + cell
        int cellpos = ci - b * R2_;
        atomicAdd(&out[(((size_t)k * B_ + b) * H_ + ch) * R2_ + cellpos], v);
        if (ch == 0) atomicAdd(&cnt[(size_t)k * CELLS_ + ci], 1.0f);
    }
}

__global__ void norm_kernel(float* __restrict__ out, const float* __restrict__ cnt) {
    int g = blockIdx.x * blockDim.x + threadIdx.x;
    const int total = 3 * B_ * H_ * R2_;
    if (g >= total) return;
    int cellpos = g & (R2_ - 1);
    int rest = g >> 12;        // (plane*B + b)*H + ch
    int pb = rest >> 7;        // plane*B + b
    float n = cnt[(size_t)pb * R2_ + cellpos];
    out[g] = out[g] / fmaxf(n, 1.0f);
}

// ---------------- host orchestration ----------------------------------------
extern "C" void kernel_launch(void* const* d_in, const int* in_sizes, int n_in,
                              void* d_out, int out_size, void* d_ws, size_t ws_size,
                              hipStream_t stream) {
    const float* p        = (const float*)d_in[0];
    const float* fc_pos_W = (const float*)d_in[1];
    const float* fc_pos_b = (const float*)d_in[2];
    const float* W0       = (const float*)d_in[3];
    const float* b0       = (const float*)d_in[4];
    const float* W1       = (const float*)d_in[5];
    const float* b1       = (const float*)d_in[6];
    const float* Ws       = (const float*)d_in[7];
    const float* fc_c_W   = (const float*)d_in[8];
    const float* fc_c_b   = (const float*)d_in[9];
    float* out = (float*)d_out;

    // workspace carve (all sizes multiples of 4 KiB)
    char* w = (char*)d_ws;
    float*    netA    = (float*)w;    w += (size_t)N_ * H_ * 4;          // 64 MiB
    float*    netB    = (float*)w;    w += (size_t)N_ * H_ * 4;          // 64 MiB
    unsigned* cell    = (unsigned*)w; w += (size_t)3 * CELLS_ * H_ * 4;  // 96 MiB
    int*      idx     = (int*)w;      w += (size_t)3 * N_ * 4;           // 1.5 MiB
    float*    meanbuf = (float*)w;    w += (size_t)B_ * H_ * 4;          // 8 KiB
    float*    cnt     = (float*)w;    w += (size_t)3 * CELLS_ * 4;       // 768 KiB

    const dim3 blk(256);
    const size_t rb_shmem  = (size_t)(ROWS_ * LDX_ + ROWS_ * LDT_) * sizeof(float);
    const size_t fcc_shmem = (size_t)(ROWS_ * LDT_) * sizeof(float);

    idx_kernel<<<(N_ + 255) / 256, blk, 0, stream>>>(p, idx);

    // block 0 (input = fc_pos(p), built in LDS)
    resblock_kernel<true><<<N_ / ROWS_, blk, rb_shmem, stream>>>(
        p, fc_pos_W, fc_pos_b, nullptr, nullptr, nullptr,
        W0, b0, W1, b1, Ws, netA);

    float* cur = netA; float* nxt = netB;
    for (int i = 1; i < NB_; ++i) {
        zero_kernel<<<4096, blk, 0, stream>>>(cell, (long long)3 * CELLS_ * H_);
        zero_kernel<<<8, blk, 0, stream>>>((unsigned*)meanbuf, (long long)B_ * H_);
        scatter_max_kernel<<<(N_ * H_) / 256, blk, 0, stream>>>(cur, idx, cell);
        mean_kernel<<<B_ * (T_ / 256), dim3(128), 0, stream>>>(cur, meanbuf);
        resblock_kernel<false><<<N_ / ROWS_, blk, rb_shmem, stream>>>(
            cur, nullptr, nullptr, cell, idx, meanbuf,
            W0 + (size_t)i * K3_ * H_, b0 + (size_t)i * H_,
            W1 + (size_t)i * H_ * H_, b1 + (size_t)i * H_,
            Ws + (size_t)i * K3_ * H_, nxt);
        float* tmp = cur; cur = nxt; nxt = tmp;
    }

    // c = net @ fc_c_W + fc_c_b  (cur == netA after 4 swaps; write into nxt)
    fcc_kernel<<<N_ / ROWS_, blk, fcc_shmem, stream>>>(cur, fc_c_W, fc_c_b, nxt);

    // scatter-mean into output planes (d_out is poisoned -> zero first)
    zero_kernel<<<4096, blk, 0, stream>>>((unsigned*)out, (long long)3 * B_ * H_ * R2_);
    zero_kernel<<<64, blk, 0, stream>>>((unsigned*)cnt, (long long)3 * CELLS_);
    scatter_add_kernel<<<(N_ * H_) / 256, blk, 0, stream>>>(nxt, idx, out, cnt);
    norm_kernel<<<(3 * B_ * H_ * R2_) / 256, blk, 0, stream>>>(out, cnt);
}